// VolumeRenderer_16192026706363
// MI455X (gfx1250) — compile-verified
//
#include <hip/hip_runtime.h>

// ---------------------------------------------------------------------------
// NeRF-style volume renderer for MI455X (gfx1250, wave32).
// One wave = one ray. Field evaluation (pts @ [w_sigma|w_feat], 3x4 fp32)
// runs on the matrix pipe with V_WMMA_F32_16X16X4_F32 (exact fp32; K=4 fits
// the zero-padded 3-vector inner dim). Transmittance cumprod and CDF are
// wave32 shuffle scans; coarse+fine depth merge is a stable rank-scatter
// through LDS. Kernel is HBM-bandwidth bound (~55 MB traffic -> ~2.4 us at
// 23.3 TB/s), so per-wave instruction count is the secondary objective:
// the WMMA result scatter uses arithmetic addressing (SIG/FA/FB/FC are laid
// out at stride 80 so dst = S + n*80) instead of a branchy pointer select.
// ---------------------------------------------------------------------------

namespace {
constexpr int   kNC   = 64;
constexpr int   kNF   = 16;
constexpr int   kNT   = 80;      // NC + NF
constexpr float kZN   = 2.0f;
constexpr float kZF   = 6.0f;
constexpr float kEps  = 1e-5f;
constexpr int   kRaysPerBlock = 8;   // 8 waves * 32 lanes = 256 threads
constexpr int   kSray = 928;         // floats of LDS per ray
}

typedef __attribute__((ext_vector_type(2))) float v2f;
typedef __attribute__((ext_vector_type(8))) float v8f;

__device__ __forceinline__ float zc(int i) {
  // coarse depth i: 2 + 4*i/63  (linspace(0,1,64))
  return kZN + (kZF - kZN) * ((float)i * (1.0f / 63.0f));
}

__device__ __forceinline__ float sigmoidf(float x) {
  return 1.0f / (1.0f + expf(-x));
}

__device__ __forceinline__ float waveInclProd(float x, int lane) {
#pragma unroll
  for (int d = 1; d < 32; d <<= 1) {
    float y = __shfl_up(x, d, 32);
    if (lane >= d) x *= y;
  }
  return x;
}

__device__ __forceinline__ float waveInclSum(float x, int lane) {
#pragma unroll
  for (int d = 1; d < 32; d <<= 1) {
    float y = __shfl_up(x, d, 32);
    if (lane >= d) x += y;
  }
  return x;
}

__device__ __forceinline__ float waveSum(float x) {
#pragma unroll
  for (int d = 16; d > 0; d >>= 1) x += __shfl_xor(x, d, 32);
  return x;
}

__global__ __launch_bounds__(256)
void nerf_wmma_render(const float* __restrict__ ros, const float* __restrict__ rds,
                      const float* __restrict__ wsig, const float* __restrict__ wfeat,
                      const float* __restrict__ u, float* __restrict__ out, int nrays) {
  __shared__ float sm[kRaysPerBlock * kSray];

  const int lane = threadIdx.x & 31;
  const int wv   = threadIdx.x >> 5;
  int ray = blockIdx.x * kRaysPerBlock + wv;
  if (ray >= nrays) ray = nrays - 1;   // tail waves duplicate last ray (benign)

  float* S   = sm + wv * kSray;
  // Column arrays at stride 80 so a WMMA D column n scatters to S + n*80.
  float* SIG = S;           // [80] raw sigma (coarse 0..63, fine 64..79)
  float* FA  = S + 80;      // [80] raw feat col 0 (pre-sigmoid)
  float* FB  = S + 160;     // [80] raw feat col 1
  float* FC  = S + 240;     // [80] raw feat col 2
  float* ZAL = S + 320;     // [80] unsorted depths
  float* ZS  = S + 400;     // [80] sorted depths
  float* SS  = S + 480;     // [80] sorted sigma
  float* G0  = S + 560;     // [80] sorted feat cols
  float* G1  = S + 640;
  float* G2  = S + 720;
  float* WC  = S + 800;     // [64] coarse weights
  float* CDF = S + 864;     // [63] cdf for sample_pdf

  const float ro0 = ros[ray * 3 + 0], ro1 = ros[ray * 3 + 1], ro2 = ros[ray * 3 + 2];
  const float rd0 = rds[ray * 3 + 0], rd1 = rds[ray * 3 + 1], rd2 = rds[ray * 3 + 2];

  const int n   = lane & 15;   // WMMA column / point-within-tile
  const int hiK = lane >> 4;   // 0 -> A holds K={0,1}, 1 -> K={2,3}

  // Scatter target for this lane's D column (valid when n < 4):
  // column 0 -> SIG, 1 -> FA, 2 -> FB, 3 -> FC; rows M = v + 8*hiK.
  float* const colDst = S + n * 80 + (hiK ? 8 : 0);
  const bool   doCol  = (n < 4);

  // -------- constant B operand (4x16): col0 = w_sigma, cols1..3 = w_feat ----
  v2f bmat;
  {
    const int k0 = hiK ? 2 : 0;
    const int k1 = hiK ? 3 : 1;
    auto bval = [&](int k) -> float {
      if (k >= 3 || n >= 4) return 0.0f;                  // zero padding
      return (n == 0) ? wsig[k] : wfeat[k * 3 + (n - 1)];
    };
    bmat.x = bval(k0);
    bmat.y = bval(k1);
  }

  // -------- coarse field eval: 4 fp32 WMMA tiles of 16 points each ---------
#pragma unroll
  for (int g = 0; g < 4; ++g) {
    const float z = zc(g * 16 + n);   // A row m = n, point index g*16+n
    v2f a;
    if (hiK == 0) { a.x = fmaf(rd0, z, ro0); a.y = fmaf(rd1, z, ro1); }
    else          { a.x = fmaf(rd2, z, ro2); a.y = 0.0f; }   // K=3 pad
    v8f c = {0.f, 0.f, 0.f, 0.f, 0.f, 0.f, 0.f, 0.f};
    v8f d = __builtin_amdgcn_wmma_f32_16x16x4_f32(false, a, false, bmat,
                                                  (short)0, c, false, false);
    // D[M][N]: lane with N=n holds rows M = v (+8 if hiK) in d[v].
    if (doCol) {
#pragma unroll
      for (int v = 0; v < 8; ++v) colDst[g * 16 + v] = d[v];
    }
  }
  ZAL[lane]      = zc(lane);
  ZAL[lane + 32] = zc(lane + 32);
  __syncthreads();

  // -------- coarse transmittance / weights (uniform dist = 4/63) ----------
  {
    const float distC = (kZF - kZN) / 63.0f;
    const int i0 = 2 * lane, i1 = 2 * lane + 1;
    const float a0 = 1.0f - expf(-fmaxf(SIG[i0], 0.0f) * distC);
    const float a1 = 1.0f - expf(-fmaxf(SIG[i1], 0.0f) * distC);
    const float t0 = 1.0f - a0 + 1e-10f;
    const float t1 = 1.0f - a1 + 1e-10f;
    const float incl = waveInclProd(t0 * t1, lane);
    float excl = __shfl_up(incl, 1, 32);
    if (lane == 0) excl = 1.0f;
    WC[i0] = a0 * excl;
    WC[i1] = a1 * (excl * t0);
  }
  __syncthreads();

  // -------- sample_pdf: pdf over WC[1..62], cdf[0..62] ---------------------
  {
    float x0 = 0.0f, x1 = 0.0f;
    if (lane < 31) { x0 = WC[1 + 2 * lane] + kEps; x1 = WC[2 + 2 * lane] + kEps; }
    const float ssum = waveInclSum(x0 + x1, lane);
    const float tot  = __shfl(ssum, 31, 32);
    if (lane == 0) CDF[0] = 0.0f;
    if (lane < 31) {
      CDF[1 + 2 * lane] = (ssum - x1) / tot;
      CDF[2 + 2 * lane] = ssum / tot;
    }
  }
  __syncthreads();

  // -------- searchsorted + inverse-CDF fine depth (lanes 16..31 duplicate) -
  const float uv = u[(size_t)ray * kNF + n];
  int lo = 0, hi = 63;           // first i in [0,63) with cdf[i] >= u, else 63
#pragma unroll
  for (int it = 0; it < 6; ++it) {
    if (lo < hi) {
      const int mid = (lo + hi) >> 1;
      if (CDF[mid] < uv) lo = mid + 1; else hi = mid;
    }
  }
  const int below = (lo - 1 < 0) ? 0 : lo - 1;
  const int above = (lo < 62) ? lo : 62;
  const float c0 = CDF[below], c1 = CDF[above];
  const float b0 = 0.5f * (zc(below) + zc(below + 1));   // bins = midpoints
  const float b1 = 0.5f * (zc(above) + zc(above + 1));
  float den = c1 - c0;
  if (den < kEps) den = 1.0f;
  const float tf = b0 + (uv - c0) / den * (b1 - b0);
  if (lane < 16) ZAL[64 + n] = tf;

  // -------- fine field eval: one fp32 WMMA tile of 16 points ---------------
  {
    v2f a;
    if (hiK == 0) { a.x = fmaf(rd0, tf, ro0); a.y = fmaf(rd1, tf, ro1); }
    else          { a.x = fmaf(rd2, tf, ro2); a.y = 0.0f; }
    v8f c = {0.f, 0.f, 0.f, 0.f, 0.f, 0.f, 0.f, 0.f};
    v8f d = __builtin_amdgcn_wmma_f32_16x16x4_f32(false, a, false, bmat,
                                                  (short)0, c, false, false);
    if (doCol) {
#pragma unroll
      for (int v = 0; v < 8; ++v) colDst[64 + v] = d[v];
    }
  }
  __syncthreads();

  // -------- stable merge of coarse(sorted) + fine via rank scatter ---------
  {
    // fine rank: stable position among fine + #{coarse <= z} (coarse wins ties)
    int posF = 0;
#pragma unroll
    for (int k = 0; k < 16; ++k) {
      const float zk = ZAL[64 + k];
      posF += (zk < tf || (zk == tf && k < n)) ? 1 : 0;
    }
    int cntC = 0;
    for (int i = 0; i < 64; ++i) cntC += (zc(i) <= tf) ? 1 : 0;
    const int rankF = posF + cntC;
    if (lane < 16) {
      ZS[rankF] = tf;          SS[rankF] = SIG[64 + n];
      G0[rankF] = FA[64 + n];  G1[rankF] = FB[64 + n];  G2[rankF] = FC[64 + n];
    }
    // coarse rank: i + #{fine strictly < z_i}
#pragma unroll
    for (int h = 0; h < 2; ++h) {
      const int i  = lane + 32 * h;
      const float zi = zc(i);
      int cf = 0;
#pragma unroll
      for (int k = 0; k < 16; ++k) cf += (ZAL[64 + k] < zi) ? 1 : 0;
      const int rk = i + cf;
      ZS[rk] = zi;      SS[rk] = SIG[i];
      G0[rk] = FA[i];   G1[rk] = FB[i];   G2[rk] = FC[i];
    }
  }
  __syncthreads();

  // -------- final volume integral over 80 sorted samples -------------------
  float accR0 = 0.f, accR1 = 0.f, accR2 = 0.f, accD = 0.f;
  float tA[3], alA[3];
  float prod = 1.0f;
  const int pbase = 3 * lane;   // lanes 0..26 hold up to 3 consecutive samples
#pragma unroll
  for (int e = 0; e < 3; ++e) {
    const int p = pbase + e;
    float t = 1.0f, al = 0.0f;
    if (p < kNT) {
      const float dd = (p == kNT - 1) ? (ZS[kNT - 1] - ZS[kNT - 2])
                                      : (ZS[p + 1] - ZS[p]);
      al = 1.0f - expf(-fmaxf(SS[p], 0.0f) * dd);
      t  = 1.0f - al + 1e-10f;
    }
    tA[e] = t; alA[e] = al;
    prod *= t;
  }
  const float incl = waveInclProd(prod, lane);
  float T = __shfl_up(incl, 1, 32);
  if (lane == 0) T = 1.0f;

  float* outW = out + (size_t)nrays * 4 + (size_t)ray * kNT;
#pragma unroll
  for (int e = 0; e < 3; ++e) {
    const int p = pbase + e;
    if (p < kNT) {
      const float w = alA[e] * T;
      outW[p] = w;
      accR0 += w * sigmoidf(G0[p]);
      accR1 += w * sigmoidf(G1[p]);
      accR2 += w * sigmoidf(G2[p]);
      accD  += w * ZS[p];
    }
    T *= tA[e];
  }
  accR0 = waveSum(accR0);
  accR1 = waveSum(accR1);
  accR2 = waveSum(accR2);
  accD  = waveSum(accD);
  if (lane == 0) {
    out[(size_t)ray * 3 + 0] = accR0;
    out[(size_t)ray * 3 + 1] = accR1;
    out[(size_t)ray * 3 + 2] = accR2;
    out[(size_t)nrays * 3 + ray] = accD;
  }
}

extern "C" void kernel_launch(void* const* d_in, const int* in_sizes, int n_in,
                              void* d_out, int out_size, void* d_ws, size_t ws_size,
                              hipStream_t stream) {
  const float* ros   = (const float*)d_in[0];
  const float* rds   = (const float*)d_in[1];
  const float* wsig  = (const float*)d_in[2];
  const float* wfeat = (const float*)d_in[3];
  const float* u     = (const float*)d_in[4];
  float* out = (float*)d_out;

  const int nrays  = in_sizes[0] / 3;          // B*R
  const int blocks = (nrays + kRaysPerBlock - 1) / kRaysPerBlock;
  hipLaunchKernelGGL(nerf_wmma_render, dim3(blocks), dim3(256), 0, stream,
                     ros, rds, wsig, wfeat, u, out, nrays);
}